// sampler2d_59330678227020
// MI455X (gfx1250) — compile-verified
//
#include <hip/hip_runtime.h>
#include <math.h>

#define N_THETA 256
#define N_PHI   512
#define MPIX    (N_THETA * N_PHI)   /* 131072 */
#define BATCH   8
#define NPTS    1000000
#define RAD     8                   /* int(4.0*2 + 0.5) = 8 */
#define KSIZE   17
#define SIGMA   2.0

typedef float v2f __attribute__((ext_vector_type(2)));
typedef float v8f __attribute__((ext_vector_type(8)));

__device__ __forceinline__ int reflect_idx(int i, int n) {
  // scipy mode='reflect' == numpy 'symmetric' : -1 -> 0, n -> n-1
  if (i < 0)  i = -i - 1;
  if (i >= n) i = 2 * n - 1 - i;
  return i;
}

__global__ void zero_f32(float* __restrict__ p, int n) {
  int i = blockIdx.x * blockDim.x + threadIdx.x;
  int stride = gridDim.x * blockDim.x;
  for (; i < n; i += stride) p[i] = 0.0f;
}

// Scatter-add sums and counts into per-batch 256x512 histograms.
__global__ void scatter_kernel(const float* __restrict__ theta,
                               const float* __restrict__ phi,
                               const float* __restrict__ rm,
                               float* __restrict__ sum,
                               float* __restrict__ cnt) {
  const float PI_F = 3.14159265358979323846f;
  long total = (long)BATCH * NPTS;
  long i = (long)blockIdx.x * blockDim.x + threadIdx.x;
  long stride = (long)gridDim.x * blockDim.x;
  for (; i < total; i += stride) {
    int b = (int)(i / NPTS);
    float th = theta[i];
    float ph = phi[i];
    int tb = (int)(th / PI_F * (float)N_THETA);           // trunc toward zero
    tb = min(max(tb, 0), N_THETA - 1);
    int pb = (int)(ph / (2.0f * PI_F) * (float)N_PHI);
    pb = min(max(pb, 0), N_PHI - 1);
    int flat = b * MPIX + tb * N_PHI + pb;
    atomicAdd(&sum[flat], rm[i]);
    atomicAdd(&cnt[flat], 1.0f);
  }
}

// mean = sum / max(cnt,1); also fill analytic theta/phi grids (channels 0,1).
__global__ void mean_grid_kernel(const float* __restrict__ sum,
                                 const float* __restrict__ cnt,
                                 float* __restrict__ mean,
                                 float* __restrict__ out) {
  int i = blockIdx.x * blockDim.x + threadIdx.x;
  if (i >= BATCH * MPIX) return;
  int b = i / MPIX;
  int rem = i % MPIX;
  int t = rem / N_PHI;
  int p = rem % N_PHI;
  mean[i] = sum[i] / fmaxf(cnt[i], 1.0f);
  // tc[t] = (t+0.5)*pi/256 ; pc[p] = (p+0.5)*2pi/512 = (p+0.5)*pi/256
  const float step = 3.14159265358979323846f / 256.0f;
  size_t base = (size_t)b * 3 * MPIX;
  out[base + rem]        = ((float)t + 0.5f) * step;   // channel 0: theta grid
  out[base + MPIX + rem] = ((float)p + 0.5f) * step;   // channel 1: phi grid
}

// Build dense blur operator matrices (reflect-padded Gaussian; symmetric).
__global__ void build_blur_mats(float* __restrict__ Kt, float* __restrict__ Kp) {
  int tid = blockIdx.x * blockDim.x + threadIdx.x;
  double w[KSIZE];
  double s = 0.0;
  for (int j = 0; j < KSIZE; j++) {
    double x = (double)(j - RAD) / SIGMA;
    w[j] = exp(-0.5 * x * x);
    s += w[j];
  }
  float wf[KSIZE];
  for (int j = 0; j < KSIZE; j++) wf[j] = (float)(w[j] / s);

  if (tid < N_THETA) {
    float* row = Kt + tid * N_THETA;
    for (int j = 0; j < N_THETA; j++) row[j] = 0.0f;
    for (int j = 0; j < KSIZE; j++)
      row[reflect_idx(tid + j - RAD, N_THETA)] += wf[j];
  } else if (tid < N_THETA + N_PHI) {
    int p = tid - N_THETA;
    float* row = Kp + p * N_PHI;
    for (int j = 0; j < N_PHI; j++) row[j] = 0.0f;
    for (int j = 0; j < KSIZE; j++)
      row[reflect_idx(p + j - RAD, N_PHI)] += wf[j];
  }
}

// Generic WMMA f32 GEMM tile kernel: one 16x16 D tile per wave, K step 4.
// A-matrix layout (V_WMMA_F32_16X16X4_F32, 16x4): lanes 0-15 hold row=lane,
// K={k,k+1}; lanes 16-31 hold row=lane-16, K={k+2,k+3}.
// B-matrix (4x16): one row striped across lanes within a VGPR; lanes 0-15
// carry the first K pair, lanes 16-31 the second.
// C/D (16x16): VGPR i holds M=i (lanes 0-15) / M=i+8 (lanes 16-31), N=lane&15.
__global__ __launch_bounds__(256)
void wmma_blur(const float* __restrict__ Amat, long a_bstride, int lda,
               const float* __restrict__ Bmat, long b_bstride, int ldb,
               float* __restrict__ Dmat, long d_bstride, int ldd,
               int Kdim) {
  int wid  = (int)((blockIdx.x * blockDim.x + threadIdx.x) >> 5);
  int lane = threadIdx.x & 31;
  const int tilesN = N_PHI / 16;                       // 32
  const int tilesPerBatch = (N_THETA / 16) * tilesN;   // 512
  if (wid >= BATCH * tilesPerBatch) return;            // whole-wave uniform
  int b  = wid / tilesPerBatch;
  int r  = wid % tilesPerBatch;
  int tm = r / tilesN;
  int tn = r % tilesN;

  const float* A  = Amat + (long)b * a_bstride;
  const float* Bp = Bmat + (long)b * b_bstride;
  float*       Dp = Dmat + (long)b * d_bstride;

  int l15  = lane & 15;
  int h2   = (lane >> 4) << 1;          // 0 or 2: which K pair this lane holds
  int arow = tm * 16 + l15;
  int col  = tn * 16 + l15;

  v8f c = {};
  for (int k = 0; k < Kdim; k += 4) {
    v2f a, bb;
    a.x  = A[arow * lda + k + h2];
    a.y  = A[arow * lda + k + h2 + 1];
    bb.x = Bp[(k + h2) * ldb + col];
    bb.y = Bp[(k + h2 + 1) * ldb + col];
    c = __builtin_amdgcn_wmma_f32_16x16x4_f32(
        /*neg_a=*/false, a, /*neg_b=*/false, bb,
        /*c_mod=*/(short)0, c, /*reuse_a=*/false, /*reuse_b=*/false);
  }

  int half8 = (lane >> 4) << 3;
#pragma unroll
  for (int i = 0; i < 8; i++)
    Dp[(tm * 16 + half8 + i) * ldd + col] = c[i];
}

extern "C" void kernel_launch(void* const* d_in, const int* in_sizes, int n_in,
                              void* d_out, int out_size, void* d_ws, size_t ws_size,
                              hipStream_t stream) {
  const float* theta = (const float*)d_in[0];
  const float* phi   = (const float*)d_in[1];
  const float* rm    = (const float*)d_in[2];
  float* out = (float*)d_out;
  float* ws  = (float*)d_ws;

  const long BM = (long)BATCH * MPIX;   // 1,048,576
  float* sum  = ws;                     // [BM]
  float* cnt  = sum + BM;               // [BM]
  float* mean = cnt + BM;               // [BM]
  float* tmp  = mean + BM;              // [BM]  (theta-blurred)
  float* Kt   = tmp + BM;               // [256*256]
  float* Kp   = Kt + N_THETA * N_THETA; // [512*512]

  zero_f32<<<2048, 256, 0, stream>>>(sum, (int)(2 * BM));
  build_blur_mats<<<3, 256, 0, stream>>>(Kt, Kp);
  scatter_kernel<<<8192, 256, 0, stream>>>(theta, phi, rm, sum, cnt);
  mean_grid_kernel<<<(int)((BATCH * MPIX + 255) / 256), 256, 0, stream>>>(sum, cnt, mean, out);

  // theta pass: tmp(b) = Kt @ mean(b)        (K = 256)
  wmma_blur<<<512, 256, 0, stream>>>(Kt, 0L, N_THETA,
                                     mean, (long)MPIX, N_PHI,
                                     tmp, (long)MPIX, N_PHI, N_THETA);
  // phi pass: out_chan2(b) = tmp(b) @ Kp     (K = 512; Kp symmetric)
  wmma_blur<<<512, 256, 0, stream>>>(tmp, (long)MPIX, N_PHI,
                                     Kp, 0L, N_PHI,
                                     out + 2 * MPIX, (long)3 * MPIX, N_PHI, N_PHI);
}